// get_occu_mask_backward_1726576857613
// MI455X (gfx1250) — compile-verified
//
#include <hip/hip_runtime.h>

// Problem constants (match the reference).
namespace {
constexpr int kH  = 512;
constexpr int kW  = 1024;
constexpr int kB  = 16;
constexpr int kHW = kH * kW;        // 524288 = 2^19
constexpr int kN  = kB * kHW;       // 8388608
constexpr float kTh = 0.95f;
}  // namespace

// ---------------------------------------------------------------------------
// Kernel 1: zero the accumulation map (B128 stores, 4 floats/thread).
// ---------------------------------------------------------------------------
__global__ __launch_bounds__(256) void zero_map_kernel(float* __restrict__ map) {
  const int i = (blockIdx.x * 256 + threadIdx.x) * 4;
  *reinterpret_cast<float4*>(map + i) = float4{0.f, 0.f, 0.f, 0.f};
}

// ---------------------------------------------------------------------------
// Kernel 2: bilinear forward-splat scatter-add.
// 4 pixels per thread; flow planes loaded as float4 (GLOBAL_LOAD_B128).
// Validity gating (corner integer in range) is bit-identical to the
// reference's clip + zero-if-invalid logic.
// Atomics: no-return hardware global_atomic_add_f32 into L2-resident map.
// ---------------------------------------------------------------------------
__global__ __launch_bounds__(256) void scatter_kernel(const float* __restrict__ flow,
                                                      float* __restrict__ map) {
  const int g   = blockIdx.x * 256 + threadIdx.x;  // group of 4 pixels
  const int idx = g << 2;                          // first pixel index
  const int b   = idx >> 19;                       // kHW = 2^19
  const int p   = idx & (kHW - 1);
  const int h   = p >> 10;                         // kW = 2^10
  const int w   = p & (kW - 1);                    // multiple of 4, same row

  const float* fb = flow + (size_t)b * 2 * kHW;
  const float4 fy4 = *reinterpret_cast<const float4*>(fb + p);        // channel 0 (y-flow)
  const float4 fx4 = *reinterpret_cast<const float4*>(fb + kHW + p);  // channel 1 (x-flow)
  float* mb = map + (size_t)b * kHW;

  const float fy[4] = {fy4.x, fy4.y, fy4.z, fy4.w};
  const float fx[4] = {fx4.x, fx4.y, fx4.z, fx4.w};

#pragma unroll
  for (int k = 0; k < 4; ++k) {
    const float x = (float)(w + k) + fx[k];
    const float y = (float)h + fy[k];

    const float x1 = floorf(x);      // floor x
    const float y1 = floorf(y);      // floor y
    const float x0 = x1 + 1.f;       // ceil x
    const float y0 = y1 + 1.f;       // ceil y

    // Weights with unclipped corners (identical to ref when corner valid;
    // invalid corners are skipped entirely, matching ref's where(invalid,0)).
    const float wxf = 1.f - fabsf(x - x1);
    const float wxc = 1.f - fabsf(x - x0);
    const float wyf = 1.f - fabsf(y - y1);
    const float wyc = 1.f - fabsf(y - y0);

    const bool xfv = (x1 >= 0.f) & (x1 <= (float)(kW - 1));
    const bool xcv = (x0 >= 0.f) & (x0 <= (float)(kW - 1));
    const bool yfv = (y1 >= 0.f) & (y1 <= (float)(kH - 1));
    const bool ycv = (y0 >= 0.f) & (y0 <= (float)(kH - 1));

    const int xif = (int)x1;
    const int xic = (int)x0;
    const int yif = (int)y1;
    const int yic = (int)y0;

    if (xcv & ycv) unsafeAtomicAdd(mb + yic * kW + xic, wxc * wyc);
    if (xcv & yfv) unsafeAtomicAdd(mb + yif * kW + xic, wxc * wyf);
    if (xfv & ycv) unsafeAtomicAdd(mb + yic * kW + xif, wxf * wyc);
    if (xfv & yfv) unsafeAtomicAdd(mb + yif * kW + xif, wxf * wyf);
  }
}

// ---------------------------------------------------------------------------
// Kernel 3: threshold map -> mask (float4 in / float4 out).
// ---------------------------------------------------------------------------
__global__ __launch_bounds__(256) void mask_kernel(const float* __restrict__ map,
                                                   float* __restrict__ mask) {
  const int i = (blockIdx.x * 256 + threadIdx.x) * 4;
  const float4 v = *reinterpret_cast<const float4*>(map + i);
  float4 m;
  m.x = (v.x > kTh) ? 1.f : 0.f;
  m.y = (v.y > kTh) ? 1.f : 0.f;
  m.z = (v.z > kTh) ? 1.f : 0.f;
  m.w = (v.w > kTh) ? 1.f : 0.f;
  *reinterpret_cast<float4*>(mask + i) = m;
}

// ---------------------------------------------------------------------------
// Host launch: d_out = [occu_mask (N floats) | occu_map (N floats)].
// Inputs: d_in[0] = flow (B,2,H,W) f32; d_in[1] = grid (unused: grid
// values are exactly h,w which we regenerate bit-identically).
// ---------------------------------------------------------------------------
extern "C" void kernel_launch(void* const* d_in, const int* in_sizes, int n_in,
                              void* d_out, int out_size, void* d_ws, size_t ws_size,
                              hipStream_t stream) {
  (void)in_sizes; (void)n_in; (void)out_size; (void)d_ws; (void)ws_size;

  const float* flow = (const float*)d_in[0];
  float* mask = (float*)d_out;       // output 0: occu_mask, N floats
  float* map  = mask + kN;           // output 1: occu_map,  N floats

  const int threads = 256;                 // 8 wave32 waves per block
  const int blocks4 = kN / 4 / threads;    // 4 floats per thread -> 8192 blocks

  zero_map_kernel<<<blocks4, threads, 0, stream>>>(map);
  scatter_kernel<<<blocks4, threads, 0, stream>>>(flow, map);
  mask_kernel<<<blocks4, threads, 0, stream>>>(map, mask);
}